// StructureFeature_80582176407649
// MI455X (gfx1250) — compile-verified
//
#include <hip/hip_runtime.h>
#include <math.h>
#include <stdint.h>

// Problem constants (reference: x = [8, 32, 384, 768] f32)
#define C_   32
#define H_   384
#define W_   768
#define HW_  (H_ * W_)

// Persistent column-strip tiling with an LDS row ring buffer.
//  - 64-wide output strip, 2 pixels per thread -> 128 threads (4 waves).
//  - Ring of 10 rows/channel: 6 rows in use per band + 4 rows in flight.
//  - Each block marches 8 bands of 4 rows (32 rows per segment).
#define TX       64
#define HXX      72             // halo cols: x0-4 .. x0+67 (16B chunks, both sides aligned)
#define RROWS    10             // ring rows per channel plane
#define CPLANE   (RROWS * HXX)  // 720 floats per channel plane in LDS
#define CHUNKS_X (HXX / 4)      // 18 b128 chunks per row
#define BAND     4              // output rows per band
#define SEG      8              // bands per block
#define SEGROWS  (BAND * SEG)   // 32 output rows per block
#define NTHREADS 128

// Wave-relative LDS byte address (flat shared address, low 32 bits).
static __device__ __forceinline__ unsigned lds_addr32(const void* p) {
    return (unsigned)(uintptr_t)p;
}

// Async-copy NR halo rows (global rows g0..g0+NR-1) into ring slots
// slot0..slot0+NR-1 (mod RROWS) via the CDNA5 async-to-LDS engine.
// OOB chunks are zero-filled through normal DS stores (disjoint LDS
// addresses from the async writes -> no ordering hazard).
template <int NR>
static __device__ __forceinline__ void async_load_rows(
    const float* __restrict__ src, float* smem,
    int g0, int slot0, int x0, unsigned smem_base, int t)
{
    constexpr int TOTAL = NR * CHUNKS_X * C_;
    for (int i = t; i < TOTAL; i += NTHREADS) {
        const int c   = i / (CHUNKS_X * NR);
        const int rem = i - c * (CHUNKS_X * NR);
        const int r   = rem / CHUNKS_X;
        const int cx  = rem - r * CHUNKS_X;
        int slot = slot0 + r; if (slot >= RROWS) slot -= RROWS;
        const int gy   = g0 + r;
        const int gx   = x0 - 4 + cx * 4;    // multiple of 4 -> 16B aligned
        const int loff = c * CPLANE + slot * HXX + cx * 4;
        if ((unsigned)gy < (unsigned)H_ && (unsigned)gx < (unsigned)W_) {
            const float* g = src + (size_t)c * HW_ + (size_t)gy * W_ + gx;
            const unsigned lds = smem_base + (unsigned)(loff * 4);
            asm volatile("global_load_async_to_lds_b128 %0, %1, off"
                         :: "v"(lds), "v"(g) : "memory");
        } else {
            *reinterpret_cast<float4*>(&smem[loff]) =
                make_float4(0.f, 0.f, 0.f, 0.f);
        }
    }
}

// Per-pixel 1/max(||f||, eps) for NR freshly loaded ring rows.
template <int NR>
static __device__ __forceinline__ void inv_rows(
    const float* smem, float* sinv, int slot0, int t)
{
    constexpr int TOT = NR * HXX;
    for (int p = t; p < TOT; p += NTHREADS) {
        const int r   = p / HXX;
        const int col = p - r * HXX;
        int slot = slot0 + r; if (slot >= RROWS) slot -= RROWS;
        const int off = slot * HXX + col;
        float s = 0.f;
        #pragma unroll
        for (int c = 0; c < C_; ++c) {
            const float f = smem[c * CPLANE + off];
            s = fmaf(f, f, s);
        }
        sinv[off] = 1.0f / fmaxf(sqrtf(s), 1e-12f);
    }
}

__global__ __launch_bounds__(NTHREADS)
void affinity3x3_kernel(const float* __restrict__ x, float* __restrict__ out) {
    __shared__ float smem[C_ * CPLANE];   // 23040 floats = 90 KB feature ring
    __shared__ float sinv[RROWS * HXX];   // 720 floats: inverse norms ring

    const int t     = threadIdx.x;
    const int x0    = blockIdx.x * TX;
    const int ybase = blockIdx.y * SEGROWS;
    const int b     = blockIdx.z;

    const float* src = x + (size_t)b * C_ * HW_;
    const unsigned smem_base = lds_addr32(smem);

    // Ring mapping: global row gy lives in slot (gy - ybase + 1) mod RROWS.
    // Preload rows ybase-1 .. ybase+4 (6 rows) -> slots 0..5.
    async_load_rows<6>(src, smem, ybase - 1, 0, x0, smem_base, t);
    asm volatile("s_wait_asynccnt 0" ::: "memory");
    __syncthreads();
    inv_rows<6>(smem, sinv, 0, t);
    __syncthreads();

    const int lx = t & 31;        // pixels x0+2*lx, x0+2*lx+1
    const int ly = t >> 5;        // 0..3 (row within band)
    const int cb = 3 + 2 * lx;    // patch base col (centers at cb+1, cb+2)

    for (int k = 0; k < SEG; ++k) {
        // Prefetch next band's 4 new rows (gy = ybase+4k+5 .. +8) into
        // ring slots (4k+6 .. 4k+9) mod 10 -- disjoint from the slots
        // (4k .. 4k+5) mod 10 that this band's dot pass reads, and fenced
        // against band k-1 readers by the previous iteration's barriers.
        if (k + 1 < SEG) {
            async_load_rows<BAND>(src, smem, ybase + BAND * k + 5,
                                  (BAND * k + 6) % RROWS, x0, smem_base, t);
        }

        // ---- dot pass for band k (overlaps the in-flight async copies) ----
        const int L  = BAND * k + ly + 1;     // ring index of center row
        const int s0 = (L - 1) % RROWS;
        const int s1 = L % RROWS;
        const int s2 = (L + 1) % RROWS;

        float a0[8], a1[8];
        #pragma unroll
        for (int n = 0; n < 8; ++n) { a0[n] = 0.f; a1[n] = 0.f; }

        const float* p0 = &smem[s0 * HXX + cb];
        const float* p1 = &smem[s1 * HXX + cb];
        const float* p2 = &smem[s2 * HXX + cb];
        #pragma unroll
        for (int c = 0; c < C_; ++c) {
            const float* q0 = p0 + c * CPLANE;
            const float* q1 = p1 + c * CPLANE;
            const float* q2 = p2 + c * CPLANE;
            const float r0a = q0[0], r0b = q0[1], r0c = q0[2], r0d = q0[3];
            const float r1a = q1[0], r1b = q1[1], r1c = q1[2], r1d = q1[3];
            const float r2a = q2[0], r2b = q2[1], r2c = q2[2], r2d = q2[3];
            // pixel 0 (center r1b): reference neighbor order (dy,dx), skip center
            a0[0] = fmaf(r1b, r0a, a0[0]); a0[1] = fmaf(r1b, r0b, a0[1]); a0[2] = fmaf(r1b, r0c, a0[2]);
            a0[3] = fmaf(r1b, r1a, a0[3]); a0[4] = fmaf(r1b, r1c, a0[4]);
            a0[5] = fmaf(r1b, r2a, a0[5]); a0[6] = fmaf(r1b, r2b, a0[6]); a0[7] = fmaf(r1b, r2c, a0[7]);
            // pixel 1 (center r1c)
            a1[0] = fmaf(r1c, r0b, a1[0]); a1[1] = fmaf(r1c, r0c, a1[1]); a1[2] = fmaf(r1c, r0d, a1[2]);
            a1[3] = fmaf(r1c, r1b, a1[3]); a1[4] = fmaf(r1c, r1d, a1[4]);
            a1[5] = fmaf(r1c, r2b, a1[5]); a1[6] = fmaf(r1c, r2c, a1[6]); a1[7] = fmaf(r1c, r2d, a1[7]);
        }

        // Inverse-norm weights for the 3x4 patch (from the sinv ring).
        float w0[4], w1[4], w2[4];
        #pragma unroll
        for (int dx = 0; dx < 4; ++dx) {
            w0[dx] = sinv[s0 * HXX + cb + dx];
            w1[dx] = sinv[s1 * HXX + cb + dx];
            w2[dx] = sinv[s2 * HXX + cb + dx];
        }
        const float ic0 = w1[1];
        const float ic1 = w1[2];

        const int gy = ybase + BAND * k + ly;
        float* ob = out + (((size_t)b * 8) * H_ + gy) * (size_t)W_ + (x0 + 2 * lx);
        // n -> (dy, patch dx) in reference order; outputs never re-read -> NT stores.
        {
            float o;
            o = fmaxf(a0[0] * ic0 * w0[0], 0.f); __builtin_nontemporal_store(o, ob + 0 * (size_t)HW_);
            o = fmaxf(a1[0] * ic1 * w0[1], 0.f); __builtin_nontemporal_store(o, ob + 0 * (size_t)HW_ + 1);
            o = fmaxf(a0[1] * ic0 * w0[1], 0.f); __builtin_nontemporal_store(o, ob + 1 * (size_t)HW_);
            o = fmaxf(a1[1] * ic1 * w0[2], 0.f); __builtin_nontemporal_store(o, ob + 1 * (size_t)HW_ + 1);
            o = fmaxf(a0[2] * ic0 * w0[2], 0.f); __builtin_nontemporal_store(o, ob + 2 * (size_t)HW_);
            o = fmaxf(a1[2] * ic1 * w0[3], 0.f); __builtin_nontemporal_store(o, ob + 2 * (size_t)HW_ + 1);
            o = fmaxf(a0[3] * ic0 * w1[0], 0.f); __builtin_nontemporal_store(o, ob + 3 * (size_t)HW_);
            o = fmaxf(a1[3] * ic1 * w1[1], 0.f); __builtin_nontemporal_store(o, ob + 3 * (size_t)HW_ + 1);
            o = fmaxf(a0[4] * ic0 * w1[2], 0.f); __builtin_nontemporal_store(o, ob + 4 * (size_t)HW_);
            o = fmaxf(a1[4] * ic1 * w1[3], 0.f); __builtin_nontemporal_store(o, ob + 4 * (size_t)HW_ + 1);
            o = fmaxf(a0[5] * ic0 * w2[0], 0.f); __builtin_nontemporal_store(o, ob + 5 * (size_t)HW_);
            o = fmaxf(a1[5] * ic1 * w2[1], 0.f); __builtin_nontemporal_store(o, ob + 5 * (size_t)HW_ + 1);
            o = fmaxf(a0[6] * ic0 * w2[1], 0.f); __builtin_nontemporal_store(o, ob + 6 * (size_t)HW_);
            o = fmaxf(a1[6] * ic1 * w2[2], 0.f); __builtin_nontemporal_store(o, ob + 6 * (size_t)HW_ + 1);
            o = fmaxf(a0[7] * ic0 * w2[2], 0.f); __builtin_nontemporal_store(o, ob + 7 * (size_t)HW_);
            o = fmaxf(a1[7] * ic1 * w2[3], 0.f); __builtin_nontemporal_store(o, ob + 7 * (size_t)HW_ + 1);
        }

        // Drain the prefetch, then compute inverse norms of the new rows.
        if (k + 1 < SEG) {
            asm volatile("s_wait_asynccnt 0" ::: "memory");
            __syncthreads();
            inv_rows<BAND>(smem, sinv, (BAND * k + 6) % RROWS, t);
            __syncthreads();
        }
    }
}

extern "C" void kernel_launch(void* const* d_in, const int* in_sizes, int n_in,
                              void* d_out, int out_size, void* d_ws, size_t ws_size,
                              hipStream_t stream) {
    const float* x = (const float*)d_in[0];
    float* out = (float*)d_out;
    const int B = in_sizes[0] / (C_ * H_ * W_);   // = 8
    dim3 grid(W_ / TX, H_ / SEGROWS, B);          // (12, 12, 8) = 1152 blocks
    dim3 block(NTHREADS);
    hipLaunchKernelGGL(affinity3x3_kernel, grid, block, 0, stream, x, out);
}